// CliffordAlgebra_89429809037620
// MI455X (gfx1250) — compile-verified
//
#include <hip/hip_runtime.h>
#include <stdint.h>

// ---------------------------------------------------------------------------
// Cl(3,1,0) geometric product via the real matrix isomorphism Cl(3,1) = M4(R):
//     out = Psi( Phi(a) * Phi(b) )
// Phi/Psi are FIXED 16x16 maps -> shared-operand GEMMs -> v_wmma_f32_16x16x4_f32.
// Still HBM-bound (192 MB @ 23.3 TB/s ~ 8.2 us floor); WMMA offloads most of
// the per-point arithmetic to the matrix pipe. Async global->LDS staging
// (ASYNCcnt) with double buffering feeds the waves.
// The isomorphism is verified against the Cayley table at COMPILE TIME
// (static_assert below) -- identical math to the jnp.einsum reference.
// ---------------------------------------------------------------------------

#define NBLADES 16
#define TILE_PTS 256            // points per tile (== block size)
#define PT_STRIDE_DW 20         // 20 dwords/point in LDS (16B aligned, padded)
#define PT_STRIDE_B  (PT_STRIDE_DW * 4)          // 80 bytes
#define TILE_BYTES   (TILE_PTS * PT_STRIDE_B)    // 20480 bytes (one operand)
#define BUF_BYTES    (2 * TILE_BYTES)            // a-tile + b-tile
#define SMEM_BYTES   (2 * BUF_BYTES)             // double buffered

typedef float v2f __attribute__((ext_vector_type(2)));
typedef float v8f __attribute__((ext_vector_type(8)));

// ---- Cayley sign for Cl(3,1,0): metric (+,+,+,-) --------------------------
constexpr int cl31_sign(int a, int b) {
    int inv = 0;
    for (int t = a >> 1; t; t >>= 1) inv += __builtin_popcount(t & b);
    int s = (inv & 1) ? -1 : 1;
    if (a & b & 8) s = -s;          // e3 (bit 3) squares to -1
    return s;
}
struct SignTab { signed char s[NBLADES][NBLADES]; };
constexpr SignTab make_signs() {
    SignTab t{};
    for (int i = 0; i < NBLADES; ++i)
        for (int j = 0; j < NBLADES; ++j)
            t.s[i][j] = (signed char)cl31_sign(i, j);
    return t;
}
constexpr SignTab kSign = make_signs();

// ---- real Majorana representation of Cl(3,1) ------------------------------
struct M2i { int v[2][2]; };
struct M4i { int v[4][4]; };
constexpr M2i SX{{{0, 1}, {1, 0}}};
constexpr M2i SZ{{{1, 0}, {0, -1}}};
constexpr M2i EP{{{0, 1}, {-1, 0}}};   // eps, eps^2 = -I
constexpr M2i I2{{{1, 0}, {0, 1}}};

constexpr M4i kron(M2i P, M2i Q) {
    M4i M{};
    for (int r1 = 0; r1 < 2; ++r1)
        for (int c1 = 0; c1 < 2; ++c1)
            for (int r0 = 0; r0 < 2; ++r0)
                for (int c0 = 0; c0 < 2; ++c0)
                    M.v[2 * r1 + r0][2 * c1 + c0] = P.v[r1][c1] * Q.v[r0][c0];
    return M;
}
constexpr M4i mmul4(M4i A, M4i B) {
    M4i C{};
    for (int r = 0; r < 4; ++r)
        for (int c = 0; c < 4; ++c) {
            int s = 0;
            for (int k = 0; k < 4; ++k) s += A.v[r][k] * B.v[k][c];
            C.v[r][c] = s;
        }
    return C;
}
// generators: g0,g1,g2 square to +I (metric +), g3 squares to -I (metric -)
constexpr M4i generator(int i) {
    return i == 0 ? kron(SX, I2)
         : i == 1 ? kron(SZ, I2)
         : i == 2 ? kron(EP, EP)
         :          kron(EP, SX);
}
constexpr M4i blade_mat(int S) {     // e_S = product of generators, ascending
    M4i M{{{1,0,0,0},{0,1,0,0},{0,0,1,0},{0,0,0,1}}};
    for (int i = 0; i < 4; ++i)
        if ((S >> i) & 1) M = mmul4(M, generator(i));
    return M;
}
struct PhiTab { int v[NBLADES][16]; };   // PHI[S][m] = blade_mat(S)[m/4][m%4]
constexpr PhiTab make_phi() {
    PhiTab t{};
    for (int S = 0; S < NBLADES; ++S) {
        M4i M = blade_mat(S);
        for (int m = 0; m < 16; ++m) t.v[S][m] = M.v[m / 4][m % 4];
    }
    return t;
}
constexpr PhiTab kPhi = make_phi();

// Compile-time proof: Psi(Phi(e_i) * Phi(e_j)) == sign(i,j) * e_{i^j}.
// (Psi coefficients: out_S = (1/4) * sum_m PHI[S][m] * C_flat[m], since the
//  blade images are signed permutation matrices, orthogonal under trace.)
constexpr bool verify_iso() {
    for (int i = 0; i < NBLADES; ++i)
        for (int j = 0; j < NBLADES; ++j) {
            M4i C = mmul4(blade_mat(i), blade_mat(j));
            for (int S = 0; S < NBLADES; ++S) {
                int coef = 0;
                for (int m = 0; m < 16; ++m) coef += kPhi.v[S][m] * C.v[m / 4][m % 4];
                int expect = (S == (i ^ j)) ? 4 * cl31_sign(i, j) : 0;
                if (coef != expect) return false;
            }
        }
    return true;
}
static_assert(verify_iso(), "Cl(3,1)->M4(R) isomorphism mismatch vs Cayley table");

// ---- per-lane constant A-operands for the two shared-matrix GEMMs ---------
// A-matrix 16x4 f32 layout (ISA 7.12.2): lane = row M (mod 16),
// VGPR w + 2*(lane>>4) = local K.  Forward A = Phi^T (row m, K = blade i);
// backward A = (1/4)Phi (row S, K = m).
struct LaneTab { float v[4][2][32]; };
constexpr LaneTab make_fwdA() {
    LaneTab t{};
    for (int s = 0; s < 4; ++s)
        for (int w = 0; w < 2; ++w)
            for (int l = 0; l < 32; ++l) {
                int m = l & 15, h = l >> 4, k = 4 * s + 2 * h + w;  // K = blade
                t.v[s][w][l] = (float)kPhi.v[k][m];                 // Phi^T[m][k]
            }
    return t;
}
constexpr LaneTab make_bwdA() {
    LaneTab t{};
    for (int s = 0; s < 4; ++s)
        for (int w = 0; w < 2; ++w)
            for (int l = 0; l < 32; ++l) {
                int S = l & 15, h = l >> 4, k = 4 * s + 2 * h + w;  // K = entry m
                t.v[s][w][l] = 0.25f * (float)kPhi.v[S][k];
            }
    return t;
}
constexpr LaneTab kFwdA = make_fwdA();
constexpr LaneTab kBwdA = make_bwdA();

// ---- CDNA5 async global->LDS copy (tracked by ASYNCcnt) --------------------
__device__ __forceinline__ void async_copy_b128(uint32_t lds_byte_addr,
                                                const float* gptr) {
    asm volatile("global_load_async_to_lds_b128 %0, %1, off"
                 :: "v"(lds_byte_addr), "v"(gptr)
                 : "memory");
}
__device__ __forceinline__ void wait_async_zero() {
    asm volatile("s_wait_asynccnt 0" ::: "memory");
}

__device__ __forceinline__ v8f wmma_f32_4(v2f A, v2f B, v8f C) {
    return __builtin_amdgcn_wmma_f32_16x16x4_f32(false, A, false, B,
                                                 (short)0, C, false, false);
}

// ---------------------------------------------------------------------------
__global__ __launch_bounds__(TILE_PTS)
void CliffordAlgebra_gp_kernel(const float* __restrict__ a,
                               const float* __restrict__ b,
                               float* __restrict__ out,
                               int nTiles) {
    __shared__ __align__(16) uint8_t smem[SMEM_BYTES];
    const int tid  = threadIdx.x;
    const int lane = tid & 31;
    const int wave = tid >> 5;
    const int pt16 = lane & 15;
    const int h    = lane >> 4;      // lane half: rows m/S 0-7 vs 8-15
    const bool hi  = (h != 0);
    const uint32_t smem_base = (uint32_t)(uintptr_t)(&smem[0]);

    // constant WMMA A-operands (loaded once; cached/broadcast)
    v2f fwdA[4], bwdA[4];
#pragma unroll
    for (int s = 0; s < 4; ++s) {
        fwdA[s] = v2f{kFwdA.v[s][0][lane], kFwdA.v[s][1][lane]};
        bwdA[s] = v2f{kBwdA.v[s][0][lane], kBwdA.v[s][1][lane]};
    }

    // async fill of one tile (a + b) into buffer `buf` (sequential 512B/wave
    // bursts from HBM, scattered into the padded per-point LDS layout)
    auto issue_tile = [&](int tile, int buf) {
        const float* ga = a + (size_t)tile * TILE_PTS * NBLADES;
        const float* gb = b + (size_t)tile * TILE_PTS * NBLADES;
        const uint32_t base = smem_base + (uint32_t)buf * BUF_BYTES;
#pragma unroll
        for (int k = 0; k < 4; ++k) {
            const int g = k * TILE_PTS + tid;   // 16B chunk id, 0..1023
            const int p = g >> 2;
            const int c = g & 3;
            const uint32_t lds_a = base + (uint32_t)(p * PT_STRIDE_B + c * 16);
            async_copy_b128(lds_a, ga + (size_t)g * 4);
            async_copy_b128(lds_a + TILE_BYTES, gb + (size_t)g * 4);
        }
    };

    const int tile0 = blockIdx.x;
    if (tile0 >= nTiles) return;
    issue_tile(tile0, 0);

    int buf = 0;
    for (int tile = tile0; tile < nTiles; tile += gridDim.x) {
        wait_async_zero();     // this wave's fills of `buf` arrived
        __syncthreads();       // everyone past buf^1 reads -> safe to refill

        const int nxt = tile + gridDim.x;
        if (nxt < nTiles) issue_tile(nxt, buf ^ 1);

        // each wave handles 2 groups of 16 points
#pragma unroll
        for (int q = 0; q < 2; ++q) {
            const int pt = wave * 32 + q * 16 + pt16;
            const uint32_t abase = (uint32_t)buf * BUF_BYTES
                                 + (uint32_t)(pt * PT_STRIDE_B + h * 8);
            const uint32_t bbase = abase + TILE_BYTES;

            // ---- forward GEMMs: A4 = Phi^T * Xa^T, B4 = Phi^T * Xb^T ------
            // B-operand straight from LDS: lane holds (point l&15,
            // blades 4s+2h, 4s+2h+1) == B VGPR layout for step s.
            v8f Am = {}, Bm = {};
#pragma unroll
            for (int s = 0; s < 4; ++s) {
                const float2 xa = *reinterpret_cast<const float2*>(&smem[abase + 16 * s]);
                Am = wmma_f32_4(fwdA[s], v2f{xa.x, xa.y}, Am);
            }
#pragma unroll
            for (int s = 0; s < 4; ++s) {
                const float2 xb = *reinterpret_cast<const float2*>(&smem[bbase + 16 * s]);
                Bm = wmma_f32_4(fwdA[s], v2f{xb.x, xb.y}, Bm);
            }

            // ---- per-point 4x4 matmul: C4 = A4 * B4 -----------------------
            // lane h holds flat entries m = 8h..8h+7 (rows 2h, 2h+1);
            // exchange halves of B4 so each lane sees the full 4x4 B.
            float own[8], sh[8];
#pragma unroll
            for (int k = 0; k < 8; ++k) {
                own[k] = Bm[k];
                sh[k]  = __shfl_xor(own[k], 16, 32);
            }
            float Bfull[16];
#pragma unroll
            for (int m = 0; m < 16; ++m)
                Bfull[m] = (m < 8) ? (hi ? sh[m] : own[m])
                                   : (hi ? own[m - 8] : sh[m - 8]);
            float cf[8];
#pragma unroll
            for (int rr = 0; rr < 2; ++rr)
#pragma unroll
                for (int c = 0; c < 4; ++c) {
                    float acc = Am[4 * rr + 0] * Bfull[0 + c];
                    acc = __builtin_fmaf(Am[4 * rr + 1], Bfull[4 + c],  acc);
                    acc = __builtin_fmaf(Am[4 * rr + 2], Bfull[8 + c],  acc);
                    acc = __builtin_fmaf(Am[4 * rr + 3], Bfull[12 + c], acc);
                    cf[4 * rr + c] = acc;
                }

            // ---- backward GEMM: out = (1/4)Phi * C4 -----------------------
            // B-operand for step s needs C4 entry m = 4s+2h+w in VGPR w.
            float s0 = __shfl_xor(cf[0], 16, 32), s1 = __shfl_xor(cf[1], 16, 32);
            float s2 = __shfl_xor(cf[2], 16, 32), s3 = __shfl_xor(cf[3], 16, 32);
            float s4 = __shfl_xor(cf[4], 16, 32), s5 = __shfl_xor(cf[5], 16, 32);
            float s6 = __shfl_xor(cf[6], 16, 32), s7 = __shfl_xor(cf[7], 16, 32);
            v2f Bop0{hi ? s2 : cf[0], hi ? s3 : cf[1]};
            v2f Bop1{hi ? s6 : cf[4], hi ? s7 : cf[5]};
            v2f Bop2{hi ? cf[2] : s0, hi ? cf[3] : s1};
            v2f Bop3{hi ? cf[6] : s4, hi ? cf[7] : s5};

            v8f O = {};
            O = wmma_f32_4(bwdA[0], Bop0, O);
            O = wmma_f32_4(bwdA[1], Bop1, O);
            O = wmma_f32_4(bwdA[2], Bop2, O);
            O = wmma_f32_4(bwdA[3], Bop3, O);

            // lane h stores point pt's blades [8h, 8h+8)
            float* po = out + ((size_t)tile * TILE_PTS + (size_t)pt) * NBLADES + h * 8;
            reinterpret_cast<float4*>(po)[0] = make_float4(O[0], O[1], O[2], O[3]);
            reinterpret_cast<float4*>(po)[1] = make_float4(O[4], O[5], O[6], O[7]);
        }
        buf ^= 1;
    }
}

// Tail kernel (direct sign-folded FMA) for remainder points; none for 1024^2.
__global__ void CliffordAlgebra_gp_tail(const float* __restrict__ a,
                                        const float* __restrict__ b,
                                        float* __restrict__ out,
                                        long long base, int rem) {
    const int t = threadIdx.x;
    if (t >= rem) return;
    const float* pa = a + (base + t) * NBLADES;
    const float* pb = b + (base + t) * NBLADES;
    float av[NBLADES], bv[NBLADES], acc[NBLADES];
#pragma unroll
    for (int k = 0; k < NBLADES; ++k) { av[k] = pa[k]; bv[k] = pb[k]; acc[k] = 0.0f; }
#pragma unroll
    for (int i = 0; i < NBLADES; ++i)
#pragma unroll
        for (int j = 0; j < NBLADES; ++j) {
            const float x = (kSign.s[i][j] > 0) ? av[i] : -av[i];
            acc[i ^ j] = __builtin_fmaf(x, bv[j], acc[i ^ j]);
        }
    float* po = out + (base + t) * NBLADES;
#pragma unroll
    for (int k = 0; k < NBLADES; ++k) po[k] = acc[k];
}

extern "C" void kernel_launch(void* const* d_in, const int* in_sizes, int n_in,
                              void* d_out, int out_size, void* d_ws, size_t ws_size,
                              hipStream_t stream) {
    const float* a = (const float*)d_in[0];
    const float* b = (const float*)d_in[1];
    float* out = (float*)d_out;

    const long long npts = (long long)in_sizes[0] / NBLADES;   // 1024*1024
    const int nTiles = (int)(npts / TILE_PTS);
    const int rem = (int)(npts % TILE_PTS);

    if (nTiles > 0) {
        int blocks = nTiles < 1024 ? nTiles : 1024;
        CliffordAlgebra_gp_kernel<<<blocks, TILE_PTS, 0, stream>>>(a, b, out, nTiles);
    }
    if (rem > 0) {
        CliffordAlgebra_gp_tail<<<1, TILE_PTS, 0, stream>>>(
            a, b, out, (long long)nTiles * TILE_PTS, rem);
    }
}